// PPScatter_80848464380139
// MI455X (gfx1250) — compile-verified
//
#include <hip/hip_runtime.h>

// Problem constants (match reference shapes)
#define B_ 4
#define C_ 64
#define P_ 12000
#define H_ 512
#define W_ 512

typedef float f32x4 __attribute__((ext_vector_type(4)));

// ---------------------------------------------------------------------------
// Kernel 1: bulk zero-fill of the 256 MB canvas via the CDNA5 async path.
// A 4KB LDS buffer of zeros is the DMA source; each lane issues
// GLOBAL_STORE_ASYNC_FROM_LDS_B128 (GV mode: 64-bit vaddr, saddr=off) with
// th:NT so the 256 MB fill (> 192 MB L2) doesn't thrash the cache. ASYNCcnt
// allows 63 outstanding per wave (HW stalls issue on overflow), so a modest
// number of waves saturates the HBM write path (~11 us at 23.3 TB/s).
// ---------------------------------------------------------------------------
__global__ void ppscatter_fill_kernel(float* __restrict__ out, long n_elems) {
    __shared__ __align__(16) float zbuf[1024];          // 4 KB of zeros
    f32x4* zb4 = (f32x4*)zbuf;
    f32x4 z = {0.0f, 0.0f, 0.0f, 0.0f};
    for (int i = threadIdx.x; i < 1024 / 4; i += blockDim.x) zb4[i] = z;
    __syncthreads();

    // LDS byte address = low 32 bits of the flat shared-aperture pointer.
    // Stripe lanes across the 4KB buffer (256 threads * 16B = 4096B).
    unsigned lds_addr = (unsigned)(unsigned long long)(&zbuf[0])
                      + ((threadIdx.x * 16u) & 4095u);

    long n4 = n_elems >> 2;
    f32x4* __restrict__ out4 = (f32x4*)out;
    long i = (long)blockIdx.x * blockDim.x + threadIdx.x;
    long stride = (long)gridDim.x * blockDim.x;
    for (; i < n4; i += stride) {
        unsigned long long gaddr = (unsigned long long)(out4 + i);
        asm volatile("global_store_async_from_lds_b128 %0, %1, off th:TH_STORE_NT"
                     :: "v"(gaddr), "v"(lds_addr) : "memory");
    }
    // Scalar tail (n_elems divisible by 4 here; keep safe + deterministic).
    if (blockIdx.x == 0 && threadIdx.x < (n_elems & 3)) {
        __builtin_nontemporal_store(0.0f, out + (n4 << 2) + threadIdx.x);
    }
    // Drain async stores (S_ENDPGM also implies wait-idle; explicit for clarity).
    asm volatile("s_wait_asynccnt 0" ::: "memory");
}

// ---------------------------------------------------------------------------
// Kernel 2: scatter. One thread per (batch, pillar). Lanes cover consecutive
// pillars, so for each channel c the wave's loads x[b][c][p..p+31] are one
// coalesced 128B transaction; the 64-deep channel loop is fully unrolled so a
// single wave keeps ~64 loads in flight. x is read exactly once ->
// non-temporal loads. inds read as int3 (global_load_b96, 384 contiguous
// bytes per wave). Output stores are random 4B -> default temporal policy.
// ---------------------------------------------------------------------------
__global__ void ppscatter_scatter_kernel(const float* __restrict__ x,
                                         const int* __restrict__ inds,
                                         float* __restrict__ out) {
    int g = blockIdx.x * blockDim.x + threadIdx.x;   // global pillar id in [0, B*P)
    if (g >= B_ * P_) return;
    int b = g / P_;
    int p = g - b * P_;

    const int* ip = inds + 3l * g;                   // inds[b][p][0..2]
    int flag = ip[0];
    int xi   = ip[1];
    int yi   = ip[2];

    const float* __restrict__ xsrc = x + ((long)b * C_) * P_ + p;  // x[b][0][p]

    // Warm the 48KB-pitch row walk (global_prefetch_b8; speculative, OOB-safe).
    __builtin_prefetch(xsrc, 0, 0);
    __builtin_prefetch(xsrc + 16l * P_, 0, 0);
    __builtin_prefetch(xsrc + 32l * P_, 0, 0);
    __builtin_prefetch(xsrc + 48l * P_, 0, 0);

    if (flag != 1) return;                           // empty pillar: dropped

    float* __restrict__ dst =
        out + (((long)b * C_) * H_ + (long)yi) * (long)W_ + (long)xi;

#pragma unroll
    for (int c = 0; c < C_; ++c) {
        float v = __builtin_nontemporal_load(xsrc + (long)c * P_);  // read-once stream
        dst[(long)c * (H_ * W_)] = v;                // out[b][c][y][x]
    }
}

extern "C" void kernel_launch(void* const* d_in, const int* in_sizes, int n_in,
                              void* d_out, int out_size, void* d_ws, size_t ws_size,
                              hipStream_t stream) {
    const float* x    = (const float*)d_in[0];   // [B, C, P] float32
    const int*   inds = (const int*)d_in[1];     // [B, P, 3] int32
    float*       out  = (float*)d_out;           // [B, C, H, W] float32

    // --- Fill: 16.7M float4 async stores; 8192 blocks x 256 threads
    // (8 wave32 per WG) -> 8 async b128 stores per thread, deeply pipelined.
    const int threads = 256;
    const int fill_blocks = 8192;
    ppscatter_fill_kernel<<<fill_blocks, threads, 0, stream>>>(out, (long)out_size);

    // --- Scatter: one thread per pillar, same stream => ordered after fill.
    const int total_pillars = B_ * P_;
    const int sc_blocks = (total_pillars + threads - 1) / threads;
    ppscatter_scatter_kernel<<<sc_blocks, threads, 0, stream>>>(x, inds, out);
}